// ScatterVertical_40656160424523
// MI455X (gfx1250) — compile-verified
//
#include <hip/hip_runtime.h>

typedef __attribute__((ext_vector_type(2))) float v2f;
typedef __attribute__((ext_vector_type(4))) float v4f;
typedef __attribute__((ext_vector_type(8))) float v8f;

#define SLOTS 9
#define IN_C 64
#define OUT_C 64
#define WAVES_PER_BLOCK 8
#define ROWS_PER_WAVE 16
#define ROWS_PER_BLOCK (WAVES_PER_BLOCK * ROWS_PER_WAVE)   // 128

// One wave computes a 16x64 output tile via v_wmma_f32_16x16x4_f32.
// Weights for the current slot (64x64 f32 = 16 KB) are staged in LDS.
__global__ __launch_bounds__(256)
void scatter_vertical_gemm(const float* __restrict__ x0, const float* __restrict__ x1,
                           const float* __restrict__ x2, const float* __restrict__ x3,
                           const float* __restrict__ x4, const float* __restrict__ x5,
                           const float* __restrict__ x6, const float* __restrict__ x7,
                           const float* __restrict__ x8,
                           const float* __restrict__ w,     // [9][64][64]
                           const float* __restrict__ bias,  // [9][64]
                           float* __restrict__ out)         // [N][64]
{
    __shared__ float wsm[OUT_C * IN_C];  // one slot's weights, 16 KB

    const float* xs[SLOTS] = {x0, x1, x2, x3, x4, x5, x6, x7, x8};

    const int tid   = threadIdx.x;
    const int wave  = tid >> 5;          // 0..7 (wave32)
    const int lane  = tid & 31;
    const int lmod  = lane & 15;         // row/col index within 16
    const int lhalf = lane >> 4;         // 0 or 1 -> K offset 0 or 2

    const long mbase = (long)blockIdx.x * ROWS_PER_BLOCK + (long)wave * ROWS_PER_WAVE;

    v8f acc0 = {}, acc1 = {}, acc2 = {}, acc3 = {};

#pragma unroll
    for (int s = 0; s < SLOTS; ++s) {
        // Stage this slot's 64x64 weight matrix into LDS (1024 float4 / 256 thr).
        __syncthreads();
        {
            const v4f* src = (const v4f*)(w + (long)s * (OUT_C * IN_C));
            v4f*       dst = (v4f*)wsm;
#pragma unroll
            for (int i = 0; i < 4; ++i)
                dst[tid + i * 256] = src[tid + i * 256];
        }
        __syncthreads();

        // Per-lane A base: row = mbase + lmod, starting K-in-block = 2*lhalf.
        const float* arow = xs[s] + (mbase + lmod) * IN_C + 2 * lhalf;
        const float* brow = wsm + 2 * lhalf;  // + (nt*16+lmod)*IN_C + c0

#pragma unroll
        for (int c0 = 0; c0 < IN_C; c0 += 4) {
            v2f a = *(const v2f*)(arow + c0);

            v2f b0 = *(const v2f*)(brow + (0 * 16 + lmod) * IN_C + c0);
            v2f b1 = *(const v2f*)(brow + (1 * 16 + lmod) * IN_C + c0);
            v2f b2 = *(const v2f*)(brow + (2 * 16 + lmod) * IN_C + c0);
            v2f b3 = *(const v2f*)(brow + (3 * 16 + lmod) * IN_C + c0);

            acc0 = __builtin_amdgcn_wmma_f32_16x16x4_f32(false, a, false, b0, (short)0, acc0, false, false);
            acc1 = __builtin_amdgcn_wmma_f32_16x16x4_f32(false, a, false, b1, (short)0, acc1, false, false);
            acc2 = __builtin_amdgcn_wmma_f32_16x16x4_f32(false, a, false, b2, (short)0, acc2, false, false);
            acc3 = __builtin_amdgcn_wmma_f32_16x16x4_f32(false, a, false, b3, (short)0, acc3, false, false);
        }
    }

    // Summed bias per output column (N = nt*16 + lmod for this lane).
    float bsum[4];
#pragma unroll
    for (int nt = 0; nt < 4; ++nt) {
        float bs = 0.0f;
#pragma unroll
        for (int k = 0; k < SLOTS; ++k)
            bs += bias[k * OUT_C + nt * 16 + lmod];
        bsum[nt] = bs;
    }

    // C/D layout: VGPR j -> M = j + 8*lhalf, N = nt*16 + lmod.
    v8f accs[4] = {acc0, acc1, acc2, acc3};
#pragma unroll
    for (int nt = 0; nt < 4; ++nt) {
#pragma unroll
        for (int j = 0; j < 8; ++j) {
            long row = mbase + j + 8 * lhalf;
            out[row * OUT_C + nt * 16 + lmod] = accs[nt][j] + bsum[nt];
        }
    }
}

extern "C" void kernel_launch(void* const* d_in, const int* in_sizes, int n_in,
                              void* d_out, int out_size, void* d_ws, size_t ws_size,
                              hipStream_t stream) {
    const float* x[SLOTS];
    for (int i = 0; i < SLOTS; ++i) x[i] = (const float*)d_in[i];
    const float* w    = (const float*)d_in[9];
    const float* bias = (const float*)d_in[10];
    float* out = (float*)d_out;

    const int nrows = in_sizes[0] / IN_C;              // 262144
    const int nblocks = (nrows + ROWS_PER_BLOCK - 1) / ROWS_PER_BLOCK;  // 2048

    scatter_vertical_gemm<<<nblocks, 256, 0, stream>>>(
        x[0], x[1], x[2], x[3], x[4], x[5], x[6], x[7], x[8], w, bias, out);
}